// StateQueue_28123445854543
// MI455X (gfx1250) — compile-verified
//
#include <hip/hip_runtime.h>
#include <stdint.h>

// ---------------------------------------------------------------------------
// StateQueue: pure memory-movement op. ~240 MB HBM traffic, 0 matrix FLOPs.
// Strategy: stream the two large tensors (mq 59MB, me 44MB) through LDS using
// CDNA5 async global<->LDS B128 DMA (ASYNCcnt), everything else direct B128.
// ---------------------------------------------------------------------------

static constexpr uint32_t Bd = 16, Nd = 900, Dd = 256, QM = 4, QP = 4, NP = 6, Td = 4;
static constexpr uint32_t DV = Dd / 4;               // float4 per row = 64

// output offsets in floats (concatenated return order)
static constexpr uint32_t SZ_mq  = Bd * QM * Nd * Dd;          // 14,745,600
static constexpr uint32_t SZ_pq  = Bd * QP * NP * Dd;          //     98,304
static constexpr uint32_t SZ_eq  = Bd * QP * Dd;               //     16,384
static constexpr uint32_t SZ_per = Bd * QM;                    //         64
static constexpr uint32_t SZ_mm  = Bd * (Td - 1) * Nd;         //     43,200
static constexpr uint32_t SZ_me  = Bd * (Td - 1) * Nd * Dd;    // 11,059,200
static constexpr uint32_t SZ_pm  = Bd * (Td - 1);              //         48
static constexpr uint32_t SZ_pe  = Bd * (Td - 1) * Dd;         //     12,288

static constexpr uint32_t O_mq  = 0;
static constexpr uint32_t O_pq  = O_mq  + SZ_mq;
static constexpr uint32_t O_eq  = O_pq  + SZ_pq;
static constexpr uint32_t O_per = O_eq  + SZ_eq;
static constexpr uint32_t O_mm  = O_per + SZ_per;
static constexpr uint32_t O_me  = O_mm  + SZ_mm;
static constexpr uint32_t O_pm  = O_me  + SZ_me;
static constexpr uint32_t O_pe  = O_pm  + SZ_pm;

// ------------------------- CDNA5 async LDS helpers -------------------------
__device__ __forceinline__ uint32_t lds_addr_of(const void* p) {
    // generic ptr to shared var: flat addr = {shared_aperture, lds_offset};
    // low 32 bits are the wave-relative LDS byte address the async ops want.
    return (uint32_t)(uintptr_t)p;
}

__device__ __forceinline__ void async_g2l_b128(uint32_t lds, const void* g) {
    asm volatile("global_load_async_to_lds_b128 %0, %1, off"
                 :: "v"(lds), "v"((uint64_t)(uintptr_t)g)
                 : "memory");
}

__device__ __forceinline__ void async_l2g_b128(void* g, uint32_t lds) {
    asm volatile("global_store_async_from_lds_b128 %0, %1, off"
                 :: "v"((uint64_t)(uintptr_t)g), "v"(lds)
                 : "memory");
}

__device__ __forceinline__ void wait_async0() {
    asm volatile("s_wait_asynccnt 0x0" ::: "memory");
}

// ------------------------------ kernel 1: mq -------------------------------
// out[b,q,n,:] = reset[b] ? motion_query[b,n,:] : motion_queue[b,q,n,:]
__global__ __launch_bounds__(256) void k_mq(const float* __restrict__ mquery,
                                            const float* __restrict__ mqueue,
                                            const unsigned char* __restrict__ mask,
                                            float* __restrict__ out) {
    __shared__ float4 stage[8 * 256];                // 32 KB staging
    const uint32_t total  = SZ_mq / 4;               // 3,686,400 float4
    const uint32_t tid    = threadIdx.x;
    const uint32_t stride = blockDim.x * gridDim.x;
    const float4* srcA = (const float4*)mquery;      // (B,N,64)
    const float4* srcB = (const float4*)mqueue;      // flat-identical to out
    float4*       dst  = (float4*)(out + O_mq);

    for (uint32_t base = blockIdx.x * blockDim.x + tid; base < total; base += stride * 8u) {
        wait_async0();                               // staging slots free (prev stores done)
#pragma unroll
        for (int k = 0; k < 8; ++k) {
            uint32_t i = base + (uint32_t)k * stride;
            if (i < total) {
                uint32_t bq  = i / (Nd * DV);        // (b*QM+q)
                uint32_t rem = i - bq * (Nd * DV);   // n*64 + d4
                uint32_t b   = bq >> 2;
                const float4* s = mask[b] ? (srcB + i)
                                          : (srcA + b * (Nd * DV) + rem);
                async_g2l_b128(lds_addr_of(&stage[k * 256 + tid]), s);
            }
        }
        wait_async0();                               // loads landed in LDS
#pragma unroll
        for (int k = 0; k < 8; ++k) {
            uint32_t i = base + (uint32_t)k * stride;
            if (i < total)
                async_l2g_b128(dst + i, lds_addr_of(&stage[k * 256 + tid]));
        }
    }
    wait_async0();
}

// ------------------------------ kernel 2: me -------------------------------
// out[b,t,n,:] = temp_anchor_embed[b,n,t,:]  for t in 0..2
__global__ __launch_bounds__(256) void k_me(const float* __restrict__ tanch,
                                            float* __restrict__ out) {
    __shared__ float4 stage[8 * 256];
    const uint32_t total  = SZ_me / 4;               // 2,764,800 float4
    const uint32_t tid    = threadIdx.x;
    const uint32_t stride = blockDim.x * gridDim.x;
    const float4* src = (const float4*)tanch;        // (B,N,T,64)
    float4*       dst = (float4*)(out + O_me);

    for (uint32_t base = blockIdx.x * blockDim.x + tid; base < total; base += stride * 8u) {
        wait_async0();
#pragma unroll
        for (int k = 0; k < 8; ++k) {
            uint32_t j = base + (uint32_t)k * stride;
            if (j < total) {
                uint32_t d4  = j & 63u;
                uint32_t row = j >> 6;               // ((b*3)+t)*900 + n
                uint32_t n   = row % Nd;
                uint32_t bt  = row / Nd;
                uint32_t t   = bt % 3u;
                uint32_t b   = bt / 3u;
                uint32_t sidx = ((b * Nd + n) * Td + t) * DV + d4;
                async_g2l_b128(lds_addr_of(&stage[k * 256 + tid]), src + sidx);
            }
        }
        wait_async0();
#pragma unroll
        for (int k = 0; k < 8; ++k) {
            uint32_t j = base + (uint32_t)k * stride;
            if (j < total)
                async_l2g_b128(dst + j, lds_addr_of(&stage[k * 256 + tid]));
        }
    }
    wait_async0();
}

// --------------------- kernel 3: pq, eq, per, mm, pm -----------------------
__global__ __launch_bounds__(256) void k_small(const float* __restrict__ pquery,
                                               const float* __restrict__ pqueue,
                                               const float* __restrict__ efeat,
                                               const float* __restrict__ equeue,
                                               const int* __restrict__ period,
                                               const unsigned char* __restrict__ mask,
                                               const unsigned char* __restrict__ tmask,
                                               const unsigned char* __restrict__ emask,
                                               float* __restrict__ out) {
    const uint32_t n_pq = SZ_pq / 4;                 // 24,576 float4
    const uint32_t n_eq = SZ_eq / 4;                 //  4,096 float4
    uint32_t g = blockIdx.x * blockDim.x + threadIdx.x;

    if (g < n_pq) {                                  // pq
        uint32_t i = g;
        uint32_t d4 = i & 63u;
        uint32_t rest = i >> 6;                      // (b*QP+q)*NP + n
        uint32_t n = rest % NP;
        uint32_t b = (rest / NP) >> 2;
        const float4* s = mask[b] ? ((const float4*)pqueue + i)
                                  : ((const float4*)pquery + (b * NP + n) * DV + d4);
        ((float4*)(out + O_pq))[i] = *s;
    } else if (g < n_pq + n_eq) {                    // eq
        uint32_t i = g - n_pq;
        uint32_t d4 = i & 63u;
        uint32_t b  = (i >> 6) >> 2;
        const float4* s = mask[b] ? ((const float4*)equeue + i)
                                  : ((const float4*)efeat + b * DV + d4);
        ((float4*)(out + O_eq))[i] = *s;
    } else if (g < n_pq + n_eq + SZ_per) {           // per (int -> exact float)
        uint32_t i = g - (n_pq + n_eq);
        uint32_t b = i >> 2;
        out[O_per + i] = mask[b] ? (float)period[i] : 0.0f;
    } else if (g < n_pq + n_eq + SZ_per + SZ_mm) {   // mm: out[b,t,n] = tmask[b,n,t]
        uint32_t i = g - (n_pq + n_eq + SZ_per);
        uint32_t n  = i % Nd;
        uint32_t bt = i / Nd;
        uint32_t t  = bt % 3u;
        uint32_t b  = bt / 3u;
        out[O_mm + i] = tmask[(b * Nd + n) * Td + t] ? 1.0f : 0.0f;
    } else if (g < n_pq + n_eq + SZ_per + SZ_mm + SZ_pm) {  // pm
        uint32_t i = g - (n_pq + n_eq + SZ_per + SZ_mm);
        uint32_t t = i % 3u;
        uint32_t b = i / 3u;
        out[O_pm + i] = emask[b * Td + t] ? 1.0f : 0.0f;
    }
}

// ------------------------------ kernel 4: pe -------------------------------
// pm[b,t] = emask[b,0,t] (t<3); any_flag = any(pm); all_true[b] = all_t pm;
// ff[b] = argmin_t pm[b,:]; repl = all_true ? anchor[b,0,3,:] : anchor[b,0,ff,:]
// pe_out[b,t,0,d] = (any_flag && (all_true[b] || t < ff[b])) ? repl[d]
//                                                            : anchor[b,0,t,d]
__global__ __launch_bounds__(256) void k_pe(const float* __restrict__ eanch,
                                            const unsigned char* __restrict__ emask,
                                            float* __restrict__ out) {
    bool any_flag = false;
    for (uint32_t b = 0; b < Bd; ++b)
        for (uint32_t t = 0; t < 3u; ++t)
            any_flag |= (emask[b * Td + t] != 0);

    for (uint32_t k = blockIdx.x * blockDim.x + threadIdx.x; k < SZ_pe;
         k += blockDim.x * gridDim.x) {
        uint32_t d  = k & 255u;
        uint32_t bt = k >> 8;
        uint32_t t  = bt % 3u;
        uint32_t b  = bt / 3u;
        bool m0 = emask[b * Td + 0] != 0;
        bool m1 = emask[b * Td + 1] != 0;
        bool m2 = emask[b * Td + 2] != 0;
        bool all_true = m0 && m1 && m2;
        uint32_t ff = m0 ? (m1 ? (m2 ? 0u : 2u) : 1u) : 0u;  // argmin (first False)
        float v;
        if (any_flag && (all_true || t < ff)) {
            uint32_t st = all_true ? 3u : ff;
            v = eanch[(b * Td + st) * Dd + d];
        } else {
            v = eanch[(b * Td + t) * Dd + d];
        }
        out[O_pe + k] = v;
    }
}

// ------------------------------- launcher ----------------------------------
extern "C" void kernel_launch(void* const* d_in, const int* in_sizes, int n_in,
                              void* d_out, int out_size, void* d_ws, size_t ws_size,
                              hipStream_t stream) {
    (void)in_sizes; (void)n_in; (void)out_size; (void)d_ws; (void)ws_size;
    const float*         motion_query = (const float*)d_in[0];
    const float*         plan_query   = (const float*)d_in[1];
    const float*         ego_feat     = (const float*)d_in[2];
    const float*         motion_queue = (const float*)d_in[3];
    const float*         plan_queue   = (const float*)d_in[4];
    const float*         ego_queue    = (const float*)d_in[5];
    const int*           period       = (const int*)d_in[6];
    const unsigned char* mask         = (const unsigned char*)d_in[7];
    const float*         tanch        = (const float*)d_in[8];
    const unsigned char* tmask        = (const unsigned char*)d_in[9];
    const float*         eanch        = (const float*)d_in[10];
    const unsigned char* emask        = (const unsigned char*)d_in[11];
    float* out = (float*)d_out;

    k_mq<<<1024, 256, 0, stream>>>(motion_query, motion_queue, mask, out);
    k_me<<<1024, 256, 0, stream>>>(tanch, out);

    const uint32_t smallTot = SZ_pq / 4 + SZ_eq / 4 + SZ_per + SZ_mm + SZ_pm; // 71,984
    k_small<<<(smallTot + 255) / 256, 256, 0, stream>>>(plan_query, plan_queue, ego_feat,
                                                        ego_queue, period, mask, tmask,
                                                        emask, out);
    k_pe<<<48, 256, 0, stream>>>(eanch, emask, out);
}